// MatchingPursuit_47313359733220
// MI455X (gfx1250) — compile-verified
//
#include <hip/hip_runtime.h>
#include <math.h>

// ---------------------------------------------------------------------------
// Basis-pursuit ADMM via Woodbury + Newton-Schulz, all GEMMs on f32 WMMA.
//   w   = normalize_rows(weight)                  [1024,512]
//   S   = I + w^T w                               [512,512]
//   Sinv via Newton-Schulz (16 iters, X0 = I/||S||_inf)
//   Minv = I - w Sinv w^T                         [1024,1024]
//   Atb  = x @ w^T                                [2048,1024]
//   100x: rhs = Atb+v-u; xk = rhs@Minv; v = st(xk+u); u = (xk+u)-v
//   out  = [ v (2048x1024) , v @ w (2048x512) ]
//
// GEMM pipeline per K-tile (double-buffered LDS):
//   A slice: GLOBAL_LOAD_ASYNC_TO_LDS_B128 (ASYNCcnt, overlapped with compute)
//   B slice: global float4 loads issued a tile ahead (latency hidden behind
//            WMMA), stored transposed [n][k] so fragments are contiguous v2f
//   compute: V_WMMA_F32_16X16X4_F32
// ---------------------------------------------------------------------------

typedef __attribute__((ext_vector_type(2))) float v2f;
typedef __attribute__((ext_vector_type(8))) float v8f;

#define LAMBD      0.2f
#define ADMM_ITERS 100
#define NS_ITERS   16

#define BATCH 2048
#define NIN   512
#define NOUT  1024

// ---- WMMA f32 GEMM: C[M,N] = alpha * (A[M,K] @ B[K,N]) + add_ident * I ----
// 128 threads = 4 waves (wave32), block tile 64x64, wave tile 32x32.
// Requires M%64==0, N%64==0, K%32==0.
#define TM 64
#define TN 64
#define TK 32
#define KPAD 4   // row stride = TK+4 floats (144B, 16B aligned)

__global__ __launch_bounds__(128) void wmma_gemm_f32(
    const float* __restrict__ A, const float* __restrict__ B,
    float* __restrict__ C, int M, int N, int K, float alpha, int add_ident)
{
  __shared__ float As[2][TM][TK + KPAD];   // [buf][m][k]
  __shared__ float Bs[2][TN][TK + KPAD];   // [buf][n][k] (transposed)

  const int tid  = threadIdx.x;
  const int lane = tid & 31;
  const int wave = tid >> 5;
  const int wm   = (wave >> 1) * 32;   // wave row offset in block tile
  const int wn   = (wave & 1) * 32;    // wave col offset
  const int l16  = lane & 15;
  const int lh   = lane >> 4;          // lane half: selects K pair / M half
  const int brow = blockIdx.y * TM;
  const int bcol = blockIdx.x * TN;

  const unsigned long long Ag = (unsigned long long)A;

  // A slice 64x32 via async global->LDS b128: 4 chunks per thread
  // => exactly 4 ASYNCcnt increments per wave per tile.
  auto stageA = [&](int k0, int buf) {
    if (k0 + TK < K) {                 // warm L2 one tile ahead of the async engine
      __builtin_prefetch(&A[(size_t)(brow + (tid >> 1)) * K + k0 + TK], 0, 1);
      __builtin_prefetch(&B[(size_t)(k0 + TK + (tid >> 2)) * N + bcol], 0, 1);
    }
    for (int c = tid; c < (TM * TK) / 4; c += 128) {
      const int r    = c >> 3;
      const int col4 = (c & 7) << 2;
      // low 32 bits of a generic LDS pointer == wave-relative LDS byte offset
      const unsigned lds  = (unsigned)(unsigned long long)&As[buf][r][col4];
      const unsigned goff = (unsigned)(((size_t)(brow + r) * K + k0 + col4) * 4u);
      asm volatile("global_load_async_to_lds_b128 %0, %1, %2"
                   :: "v"(lds), "v"(goff), "s"(Ag) : "memory");
    }
  };

  // B slice 32x64 register-pipelined: loads issued one tile ahead.
  float4 breg[4];
  auto loadB = [&](int k0) {
#pragma unroll
    for (int i = 0; i < 4; ++i) {
      const int c  = tid + i * 128;
      const int kr = c >> 4;
      const int n4 = (c & 15) << 2;
      breg[i] = *(const float4*)&B[(size_t)(k0 + kr) * N + bcol + n4];
    }
  };
  auto storeB = [&](int buf) {
#pragma unroll
    for (int i = 0; i < 4; ++i) {
      const int c  = tid + i * 128;
      const int kr = c >> 4;
      const int n4 = (c & 15) << 2;
      Bs[buf][n4 + 0][kr] = breg[i].x;
      Bs[buf][n4 + 1][kr] = breg[i].y;
      Bs[buf][n4 + 2][kr] = breg[i].z;
      Bs[buf][n4 + 3][kr] = breg[i].w;
    }
  };

  v8f acc[2][2] = {};                  // 2x2 tiles of 16x16 f32 accumulators
  const int ntiles = K / TK;

  stageA(0, 0);                        // prologue
  loadB(0);

  for (int t = 0; t < ntiles; ++t) {
    const int buf = t & 1;
    if (t + 1 < ntiles) stageA((t + 1) * TK, buf ^ 1);  // overlap A for next tile
    storeB(buf);                       // publish B(t) (regs -> LDS)
    if (t + 1 < ntiles) {
      loadB((t + 1) * TK);             // B(t+1) global loads hide behind compute
      // asyncs complete in order: <=4 outstanding => A(t) fully in LDS
      asm volatile("s_wait_asynccnt 0x4" ::: "memory");
    } else {
      asm volatile("s_wait_asynccnt 0x0" ::: "memory");
    }
    __syncthreads();

    // f32 16x4 A layout: lane<16 holds K={ka,ka+1} of row M=l16 (ka even),
    // lane>=16 same rows, K+=2.  B (4x16) mirrors it with N=l16.
#pragma unroll
    for (int kk = 0; kk < TK; kk += 4) {
      const int ka = kk + 2 * lh;
      v2f a0 = *(const v2f*)&As[buf][wm + l16][ka];
      v2f a1 = *(const v2f*)&As[buf][wm + 16 + l16][ka];
      v2f b0 = *(const v2f*)&Bs[buf][wn + l16][ka];
      v2f b1 = *(const v2f*)&Bs[buf][wn + 16 + l16][ka];
      acc[0][0] = __builtin_amdgcn_wmma_f32_16x16x4_f32(false, a0, false, b0,
                      (short)0, acc[0][0], false, false);
      acc[0][1] = __builtin_amdgcn_wmma_f32_16x16x4_f32(false, a0, false, b1,
                      (short)0, acc[0][1], false, false);
      acc[1][0] = __builtin_amdgcn_wmma_f32_16x16x4_f32(false, a1, false, b0,
                      (short)0, acc[1][0], false, false);
      acc[1][1] = __builtin_amdgcn_wmma_f32_16x16x4_f32(false, a1, false, b1,
                      (short)0, acc[1][1], false, false);
    }
    __syncthreads();   // all waves done reading `buf` before it is re-filled
  }

  // C/D layout: VGPR r -> row r (lanes 0-15) / row r+8 (lanes 16-31), col=l16
#pragma unroll
  for (int mi = 0; mi < 2; ++mi)
#pragma unroll
    for (int ni = 0; ni < 2; ++ni) {
      const int colg  = bcol + wn + ni * 16 + l16;
      const int rbase = brow + wm + mi * 16 + 8 * lh;
#pragma unroll
      for (int r = 0; r < 8; ++r) {
        const int rowg = rbase + r;
        float val = alpha * acc[mi][ni][r];
        if (add_ident && rowg == colg) val += 1.0f;
        C[(size_t)rowg * N + colg] = val;
      }
    }
}

// ---------------- elementwise / setup kernels ----------------

__global__ __launch_bounds__(128) void normalize_rows(
    const float* __restrict__ w, float* __restrict__ wn)
{
  __shared__ float red[128];
  const int row = blockIdx.x;
  const int tid = threadIdx.x;
  float s = 0.f;
  for (int c = tid; c < NIN; c += 128) {
    const float t = w[(size_t)row * NIN + c];
    s += t * t;
  }
  red[tid] = s;
  __syncthreads();
  for (int off = 64; off > 0; off >>= 1) {
    if (tid < off) red[tid] += red[tid + off];
    __syncthreads();
  }
  const float inv = rsqrtf(red[0]);
  for (int c = tid; c < NIN; c += 128)
    wn[(size_t)row * NIN + c] = w[(size_t)row * NIN + c] * inv;
}

__global__ void transpose_wn(const float* __restrict__ wn, float* __restrict__ wT)
{
  const int idx = blockIdx.x * blockDim.x + threadIdx.x;
  if (idx < NOUT * NIN) {
    const int r = idx / NIN, c = idx % NIN;
    wT[(size_t)c * NOUT + r] = wn[idx];
  }
}

__global__ __launch_bounds__(128) void row_abssum(
    const float* __restrict__ S, float* __restrict__ sums)
{
  __shared__ float red[128];
  const int row = blockIdx.x;
  const int tid = threadIdx.x;
  float s = 0.f;
  for (int c = tid; c < NIN; c += 128) s += fabsf(S[(size_t)row * NIN + c]);
  red[tid] = s;
  __syncthreads();
  for (int off = 64; off > 0; off >>= 1) {
    if (tid < off) red[tid] += red[tid + off];
    __syncthreads();
  }
  if (tid == 0) sums[row] = red[0];
}

__global__ __launch_bounds__(256) void inf_norm_inv(
    const float* __restrict__ sums, float* __restrict__ alpha)
{
  __shared__ float red[256];
  const int tid = threadIdx.x;
  red[tid] = fmaxf(sums[tid], sums[tid + 256]);
  __syncthreads();
  for (int off = 128; off > 0; off >>= 1) {
    if (tid < off) red[tid] = fmaxf(red[tid], red[tid + off]);
    __syncthreads();
  }
  if (tid == 0) alpha[0] = 1.0f / red[0];
}

__global__ void init_scaled_identity(float* __restrict__ X,
                                     const float* __restrict__ alpha, int n)
{
  const int idx = blockIdx.x * blockDim.x + threadIdx.x;
  if (idx < n * n) {
    const int r = idx / n, c = idx % n;
    X[idx] = (r == c) ? alpha[0] : 0.f;
  }
}

__global__ void two_i_minus(const float* __restrict__ T, float* __restrict__ E, int n)
{
  const int idx = blockIdx.x * blockDim.x + threadIdx.x;
  if (idx < n * n) {
    const int r = idx / n, c = idx % n;
    E[idx] = ((r == c) ? 2.f : 0.f) - T[idx];
  }
}

__global__ void zero2(float* __restrict__ a, float* __restrict__ b, int n)
{
  const int idx = blockIdx.x * blockDim.x + threadIdx.x;
  if (idx < n) { a[idx] = 0.f; b[idx] = 0.f; }
}

__global__ void compute_rhs(const float* __restrict__ Atb,
                            const float* __restrict__ v,
                            const float* __restrict__ u,
                            float* __restrict__ rhs, int n)
{
  const int idx = blockIdx.x * blockDim.x + threadIdx.x;
  if (idx < n) rhs[idx] = Atb[idx] + v[idx] - u[idx];
}

__global__ void admm_update(const float* __restrict__ x,
                            float* __restrict__ v, float* __restrict__ u, int n)
{
  const int idx = blockIdx.x * blockDim.x + threadIdx.x;
  if (idx < n) {
    const float t  = x[idx] + u[idx];
    const float vv = copysignf(fmaxf(fabsf(t) - LAMBD, 0.f), t);
    v[idx] = vv;
    u[idx] = t - vv;
  }
}

// ---------------- launch sequence ----------------

extern "C" void kernel_launch(void* const* d_in, const int* in_sizes, int n_in,
                              void* d_out, int out_size, void* d_ws, size_t ws_size,
                              hipStream_t stream)
{
  const float* x_in   = (const float*)d_in[0];   // [2048,512]
  const float* weight = (const float*)d_in[1];   // [1024,512]
  float* out = (float*)d_out;                    // [2048*1024 + 2048*512]

  float* ws = (float*)d_ws;
  size_t off = 0;
  float* wn   = ws + off; off += (size_t)NOUT * NIN;
  float* wT   = ws + off; off += (size_t)NIN * NOUT;
  float* S    = ws + off; off += (size_t)NIN * NIN;
  float* X    = ws + off; off += (size_t)NIN * NIN;
  float* Xn   = ws + off; off += (size_t)NIN * NIN;
  float* T    = ws + off; off += (size_t)NIN * NIN;
  float* E    = ws + off; off += (size_t)NIN * NIN;
  float* P    = ws + off; off += (size_t)NOUT * NIN;
  float* Minv = ws + off; off += (size_t)NOUT * NOUT;
  float* Atb  = ws + off; off += (size_t)BATCH * NOUT;
  float* v    = ws + off; off += (size_t)BATCH * NOUT;
  float* u    = ws + off; off += (size_t)BATCH * NOUT;
  float* xk   = ws + off; off += (size_t)BATCH * NOUT;
  float* rhs  = ws + off; off += (size_t)BATCH * NOUT;
  float* sums = ws + off; off += NIN;
  float* alph = ws + off; off += 1;

  auto gemm = [&](const float* A, const float* B, float* C,
                  int M, int N, int K, float alpha, int ident) {
    dim3 grid(N / TN, M / TM);
    wmma_gemm_f32<<<grid, dim3(128), 0, stream>>>(A, B, C, M, N, K, alpha, ident);
  };

  normalize_rows<<<NOUT, 128, 0, stream>>>(weight, wn);
  transpose_wn<<<(NOUT * NIN + 255) / 256, 256, 0, stream>>>(wn, wT);

  // S = I + w^T w  (512x512, K=1024)
  gemm(wT, wn, S, NIN, NIN, NOUT, 1.f, 1);

  // X0 = I / ||S||_inf, then Newton-Schulz: X <- X (2I - S X)
  row_abssum<<<NIN, 128, 0, stream>>>(S, sums);
  inf_norm_inv<<<1, 256, 0, stream>>>(sums, alph);
  init_scaled_identity<<<(NIN * NIN + 255) / 256, 256, 0, stream>>>(X, alph, NIN);
  for (int it = 0; it < NS_ITERS; ++it) {
    gemm(S, X, T, NIN, NIN, NIN, 1.f, 0);
    two_i_minus<<<(NIN * NIN + 255) / 256, 256, 0, stream>>>(T, E, NIN);
    gemm(X, E, Xn, NIN, NIN, NIN, 1.f, 0);
    float* tmp = X; X = Xn; Xn = tmp;
  }

  // Minv = I - (w Sinv) w^T   (Woodbury)
  gemm(wn, X, P, NOUT, NIN, NIN, 1.f, 0);
  gemm(P, wT, Minv, NOUT, NOUT, NIN, -1.f, 1);

  // Atb = x @ w^T
  gemm(x_in, wT, Atb, BATCH, NOUT, NIN, 1.f, 0);
  zero2<<<(BATCH * NOUT + 255) / 256, 256, 0, stream>>>(v, u, BATCH * NOUT);

  // ADMM main loop
  for (int it = 0; it < ADMM_ITERS; ++it) {
    compute_rhs<<<(BATCH * NOUT + 255) / 256, 256, 0, stream>>>(Atb, v, u, rhs,
                                                                BATCH * NOUT);
    gemm(rhs, Minv, xk, BATCH, NOUT, NOUT, 1.f, 0);
    admm_update<<<(BATCH * NOUT + 255) / 256, 256, 0, stream>>>(xk, v, u,
                                                                BATCH * NOUT);
  }

  // encoded = v ; decoded = v @ w
  hipMemcpyAsync(out, v, (size_t)BATCH * NOUT * sizeof(float),
                 hipMemcpyDeviceToDevice, stream);
  gemm(v, wn, out + (size_t)BATCH * NOUT, BATCH, NIN, NOUT, 1.f, 0);
}